// GFCNE_83863531422353
// MI455X (gfx1250) — compile-verified
//
#include <hip/hip_runtime.h>
#include <math.h>

typedef float v2f  __attribute__((ext_vector_type(2)));
typedef float v8f  __attribute__((ext_vector_type(8)));
typedef int   v4i  __attribute__((vector_size(16)));
typedef __attribute__((address_space(1))) v4i* gv4i_p;   // global int4*
typedef __attribute__((address_space(3))) v4i* lv4i_p;   // LDS int4*

#if defined(__has_builtin)
#if __has_builtin(__builtin_amdgcn_global_load_async_to_lds_b128)
#define HAVE_ASYNC_LDS 1
#endif
#endif

// ---------------------------------------------------------------- utilities
__global__ void fill_f32(float* __restrict__ p, float v, size_t n) {
  size_t i = (size_t)blockIdx.x * blockDim.x + threadIdx.x;
  if (i < n) p[i] = v;
}
__global__ void fill_u32(unsigned* __restrict__ p, unsigned v, size_t n) {
  size_t i = (size_t)blockIdx.x * blockDim.x + threadIdx.x;
  if (i < n) p[i] = v;
}

__device__ __forceinline__ unsigned ford(float f) {
  unsigned u = __float_as_uint(f);
  return (u & 0x80000000u) ? ~u : (u | 0x80000000u);
}
__device__ __forceinline__ float forddec(unsigned u) {
  unsigned v = (u & 0x80000000u) ? (u & 0x7fffffffu) : ~u;
  return __uint_as_float(v);
}
#define ORD_NEG_INF 0x007FFFFFu   // ford(-inf)

// ---------------------------------------------------------------- edge attr (Cartesian, norm=True)
__global__ void edge_absmax(const float* __restrict__ pos, const int* __restrict__ row,
                            const int* __restrict__ col, int E, unsigned* __restrict__ mx) {
  int e = blockIdx.x * blockDim.x + threadIdx.x;
  if (e >= E) return;
  int r = row[e], c = col[e];
  float dx = fabsf(pos[2 * r + 0] - pos[2 * c + 0]);
  float dy = fabsf(pos[2 * r + 1] - pos[2 * c + 1]);
  atomicMax(mx, __float_as_uint(fmaxf(dx, dy)));   // non-negative: raw bits order-preserving
}
__global__ void edge_cart(const float* __restrict__ pos, const int* __restrict__ row,
                          const int* __restrict__ col, int E, const unsigned* __restrict__ mx,
                          float* __restrict__ attr) {
  int e = blockIdx.x * blockDim.x + threadIdx.x;
  if (e >= E) return;
  int r = row[e], c = col[e];
  float s = 1.f / (2.f * __uint_as_float(mx[0]));
  attr[2 * e + 0] = (pos[2 * r + 0] - pos[2 * c + 0]) * s + 0.5f;
  attr[2 * e + 1] = (pos[2 * r + 1] - pos[2 * c + 1]) * s + 0.5f;
}

__global__ void count_idx(const int* __restrict__ idx, int n, float* __restrict__ cnt) {
  int i = blockIdx.x * blockDim.x + threadIdx.x;
  if (i < n) atomicAdd(&cnt[idx[i]], 1.f);
}
__global__ void invert_cnt(float* __restrict__ cnt, int n) {
  int i = blockIdx.x * blockDim.x + threadIdx.x;
  if (i < n) cnt[i] = 1.f / fmaxf(cnt[i], 1.f);
}

// ---------------------------------------------------------------- spline scatter (mean folded in)
// acc row stride = lda (K-padded); layout within row: kidx*Cin + channel
__global__ void scatter_spline(const float* __restrict__ x, const int* __restrict__ row,
                               const int* __restrict__ col, const float* __restrict__ attr,
                               const float* __restrict__ inv, int E, int Cin, int K2, int Kd,
                               int lda, float* __restrict__ acc) {
  int t = blockIdx.x * blockDim.x + threadIdx.x;
  int e = t / Cin, cch = t - e * Cin;
  if (e >= E) return;
  int j = row[e], i = col[e];
  float xv = x[(size_t)j * Cin + cch];
  float w0 = inv[i];
  if (K2 == 1) { atomicAdd(&acc[(size_t)i * lda + cch], w0 * xv); return; }
  float p0 = attr[2 * e + 0] * (float)(Kd - 1);
  float p1 = attr[2 * e + 1] * (float)(Kd - 1);
  float i0f = fminf(fmaxf(floorf(p0), 0.f), (float)(Kd - 2));
  float i1f = fminf(fmaxf(floorf(p1), 0.f), (float)(Kd - 2));
  float f0 = p0 - i0f, f1 = p1 - i1f;
  int i0 = (int)i0f, i1 = (int)i1f;
  size_t base = (size_t)i * lda + cch;
  float v = w0 * xv;
  atomicAdd(&acc[base + (size_t)(i0 + i1 * Kd) * Cin],           v * (1.f - f0) * (1.f - f1));
  atomicAdd(&acc[base + (size_t)(i0 + 1 + i1 * Kd) * Cin],       v * f0 * (1.f - f1));
  atomicAdd(&acc[base + (size_t)(i0 + (i1 + 1) * Kd) * Cin],     v * (1.f - f0) * f1);
  atomicAdd(&acc[base + (size_t)(i0 + 1 + (i1 + 1) * Kd) * Cin], v * f0 * f1);
}

// ---------------------------------------------------------------- zero-padded weight copy [K4][CoutP]
__global__ void pad_weights(const float* __restrict__ W, int K, int Cout, int K4, int CoutP,
                            float* __restrict__ Wp) {
  size_t t = (size_t)blockIdx.x * blockDim.x + threadIdx.x;
  if (t >= (size_t)K4 * CoutP) return;
  int r = (int)(t / CoutP), c = (int)(t - (size_t)r * CoutP);
  Wp[t] = (r < K && c < Cout) ? W[(size_t)r * Cout + c] : 0.f;
}

// ---------------------------------------------------------------- WMMA GEMM
// out = act(A[N,KA4_padded] @ WAp + X[N,KB] @ WBp + bias) (+ addin)
// KB4 > 0 : root term via WMMA (KB multiple of 4). KB4 == 0 : rank-1 root in epilogue (Cin==1).
// Weight panels (16 output columns) staged in LDS, shared by the 4 waves of the block.
__global__ void wmma_gemm(const float* __restrict__ A, const float* __restrict__ WAp, int KA4,
                          const float* __restrict__ X, const float* __restrict__ WBp, int KB,
                          int KB4, const float* __restrict__ bias, const float* __restrict__ addin,
                          float* __restrict__ out, int N, int Cout, int CoutP, int act) {
  extern __shared__ float smem[];              // [ (KA4 + KB4) * 16 ]
  const int lane = threadIdx.x;
  const int half = lane >> 4, lm = lane & 15;
  const int tileM = blockIdx.x * blockDim.y + threadIdx.y;
  const int row0 = tileM * 16, col0 = blockIdx.y * 16;

  // ---- stage weight panels into LDS (b128 chunks; CoutP is a multiple of 16 so always in-bounds)
  {
    const int tid = threadIdx.y * 32 + threadIdx.x;
    const int quads = (KA4 + KB4) * 4;
    for (int q = tid; q < quads; q += 128) {
      int r = q >> 2, p = (q & 3) * 4;
      const float* src = (r < KA4) ? (WAp + (size_t)r * CoutP + col0 + p)
                                   : (WBp + (size_t)(r - KA4) * CoutP + col0 + p);
      float* dst = &smem[r * 16 + p];
#if defined(HAVE_ASYNC_LDS)
      __builtin_amdgcn_global_load_async_to_lds_b128((gv4i_p)src, (lv4i_p)dst, 0, 0);
#else
      *(float4*)dst = *(const float4*)src;
#endif
    }
#if defined(HAVE_ASYNC_LDS)
#if defined(__has_builtin) && __has_builtin(__builtin_amdgcn_s_wait_asynccnt)
    __builtin_amdgcn_s_wait_asynccnt(0);
#else
    asm volatile("s_wait_asynccnt 0x0" ::: "memory");
#endif
#endif
    __syncthreads();
  }

  const int arow = min(row0 + lm, N - 1);      // clamp: OOB rows only affect unstored outputs
  v8f c = {0.f, 0.f, 0.f, 0.f, 0.f, 0.f, 0.f, 0.f};

  // ---- acc @ W  (KA4 is a multiple of 4; A rows are zero-padded)
  {
    const float* ap = A + (size_t)arow * KA4 + half * 2;
    for (int k0 = 0; k0 < KA4; k0 += 4) {
      v2f a = *(const v2f*)(ap + k0);
      int kb = k0 + half * 2;
      v2f b; b.x = smem[kb * 16 + lm]; b.y = smem[(kb + 1) * 16 + lm];
      c = __builtin_amdgcn_wmma_f32_16x16x4_f32(false, a, false, b, (short)0, c, false, false);
    }
  }
  // ---- x @ root (WMMA path when Cin multiple of 4)
  if (KB4 > 0) {
    const float* sB = smem + KA4 * 16;
    const float* xp = X + (size_t)arow * KB + half * 2;
    for (int k0 = 0; k0 < KB; k0 += 4) {
      v2f a = *(const v2f*)(xp + k0);
      int kb = k0 + half * 2;
      v2f b; b.x = sB[kb * 16 + lm]; b.y = sB[(kb + 1) * 16 + lm];
      c = __builtin_amdgcn_wmma_f32_16x16x4_f32(false, a, false, b, (short)0, c, false, false);
    }
  }

  // ---- epilogue
  const int n = col0 + lm;
#pragma unroll
  for (int r = 0; r < 8; ++r) {
    int m = row0 + r + half * 8;
    if (m < N && n < Cout) {
      float v = c[r] + bias[n];
      if (KB4 == 0) v += X[m] * WBp[n];        // rank-1 root term (Cin == 1)
      if (act) v = v > 0.f ? v : (__expf(v) - 1.f);   // ELU(alpha=1)
      if (addin) v += addin[(size_t)m * Cout + n];
      out[(size_t)m * Cout + n] = v;
    }
  }
}

// ---------------------------------------------------------------- batch norm (training stats)
__global__ void bn_stats(const float* __restrict__ h, int Nn, int Cc,
                         float* __restrict__ mean, float* __restrict__ istd) {
  __shared__ float ss[256], ss2[256];
  int ch = blockIdx.x;
  float s = 0.f, s2 = 0.f;
  for (int i = threadIdx.x; i < Nn; i += 256) {
    float v = h[(size_t)i * Cc + ch];
    s += v; s2 += v * v;
  }
  ss[threadIdx.x] = s; ss2[threadIdx.x] = s2;
  __syncthreads();
  for (int st = 128; st > 0; st >>= 1) {
    if (threadIdx.x < st) { ss[threadIdx.x] += ss[threadIdx.x + st]; ss2[threadIdx.x] += ss2[threadIdx.x + st]; }
    __syncthreads();
  }
  if (threadIdx.x == 0) {
    float m = ss[0] / (float)Nn;
    float var = ss2[0] / (float)Nn - m * m;
    mean[ch] = m;
    istd[ch] = rsqrtf(fmaxf(var, 0.f) + 1e-5f);
  }
}
__global__ void bn_apply(float* __restrict__ h, int Nn, int Cc,
                         const float* __restrict__ mean, const float* __restrict__ istd,
                         const float* __restrict__ g, const float* __restrict__ b) {
  size_t t = (size_t)blockIdx.x * blockDim.x + threadIdx.x;
  if (t >= (size_t)Nn * Cc) return;
  int c = (int)(t % Cc);
  h[t] = (h[t] - mean[c]) * istd[c] * g[c] + b[c];
}

// ---------------------------------------------------------------- segment ops
__global__ void seg_sum_feat(const float* __restrict__ x, const int* __restrict__ cl,
                             int n, int C, float* __restrict__ g) {
  size_t t = (size_t)blockIdx.x * blockDim.x + threadIdx.x;
  if (t >= (size_t)n * C) return;
  int i = (int)(t / C), c = (int)(t - (size_t)i * C);
  atomicAdd(&g[(size_t)cl[i] * C + c], x[t]);
}
__global__ void seg_max_feat(const float* __restrict__ x, const int* __restrict__ cl,
                             int n, int C, unsigned* __restrict__ g) {
  size_t t = (size_t)blockIdx.x * blockDim.x + threadIdx.x;
  if (t >= (size_t)n * C) return;
  int i = (int)(t / C), c = (int)(t - (size_t)i * C);
  atomicMax(&g[(size_t)cl[i] * C + c], ford(x[t]));
}
__global__ void segmax_decode(const unsigned* __restrict__ g, size_t n, float* __restrict__ out) {
  size_t t = (size_t)blockIdx.x * blockDim.x + threadIdx.x;
  if (t < n) out[t] = forddec(g[t]);
}
__global__ void pos_div(const float* __restrict__ ps, const float* __restrict__ cc,
                        int ncl, float* __restrict__ pos) {
  int t = blockIdx.x * blockDim.x + threadIdx.x;
  if (t >= ncl * 2) return;
  pos[t] = ps[t] / fmaxf(cc[t >> 1], 1.f);
}
__global__ void pweights_k(const float* __restrict__ xp, const int* __restrict__ cl,
                           const float* __restrict__ cc, const float* __restrict__ g,
                           int n, int C, float* __restrict__ wt) {
  size_t t = (size_t)blockIdx.x * blockDim.x + threadIdx.x;
  if (t >= (size_t)n * C) return;
  int i = (int)(t / C), c = (int)(t - (size_t)i * C);
  int cli = cl[i];
  float w = cc[cli] * xp[t] / (g[(size_t)cli * C + c] + 0.001f);
  wt[t] = (w != w) ? 0.f : w;
}
__global__ void upsample_mul(const float* __restrict__ hc, const int* __restrict__ cl,
                             const float* __restrict__ wt, int n, int C, float* __restrict__ out) {
  size_t t = (size_t)blockIdx.x * blockDim.x + threadIdx.x;
  if (t >= (size_t)n * C) return;
  int i = (int)(t / C), c = (int)(t - (size_t)i * C);
  out[t] = hc[(size_t)cl[i] * C + c] * wt[t];
}

// ---------------------------------------------------------------- host helpers
static inline unsigned gdiv(size_t n) { return (unsigned)((n + 255) / 256); }

static void run_conv(hipStream_t s, const float* X, int Nn, int Cin,
                     const int* ei, int E, const float* attr, const float* inv,
                     const float* Wk, int K2, int Kd,
                     const float* Rt, const float* Bs, int Cout,
                     const float* addin, float* out, int act,
                     float* acc, float* wpadA, float* wpadB) {
  int KA = K2 * Cin;
  int KA4 = (KA + 3) & ~3;
  int CoutP = ((Cout + 15) / 16) * 16;
  int KB4 = (Cin >= 4) ? Cin : 0;            // Cin is 1 or a multiple of 4
  pad_weights<<<gdiv((size_t)KA4 * CoutP), 256, 0, s>>>(Wk, KA, Cout, KA4, CoutP, wpadA);
  const float* WBp = Rt;
  if (KB4) {
    pad_weights<<<gdiv((size_t)KB4 * CoutP), 256, 0, s>>>(Rt, Cin, Cout, KB4, CoutP, wpadB);
    WBp = wpadB;
  }
  size_t an = (size_t)Nn * KA4;
  fill_f32<<<gdiv(an), 256, 0, s>>>(acc, 0.f, an);
  size_t tE = (size_t)E * Cin;
  scatter_spline<<<gdiv(tE), 256, 0, s>>>(X, ei, ei + E, attr, inv, E, Cin, K2, Kd, KA4, acc);
  dim3 gg((unsigned)(((Nn + 15) / 16 + 3) / 4), (unsigned)(CoutP / 16));
  size_t shb = (size_t)(KA4 + KB4) * 16 * sizeof(float);
  wmma_gemm<<<gg, dim3(32, 4), shb, s>>>(acc, wpadA, KA4, X, WBp, Cin, KB4, Bs, addin, out,
                                         Nn, Cout, CoutP, act);
}

extern "C" void kernel_launch(void* const* d_in, const int* in_sizes, int n_in,
                              void* d_out, int out_size, void* d_ws, size_t ws_size,
                              hipStream_t stream) {
  const float* x    = (const float*)d_in[0];
  const float* pos1 = (const float*)d_in[1];
  const int* ei[5]; for (int i = 0; i < 5; ++i) ei[i] = (const int*)d_in[2 + i];
  const int* cl[4]; for (int i = 0; i < 4; ++i) cl[i] = (const int*)d_in[7 + i];
  // conv order: 0:1a 1:1b 2:2a 3:2b 4:3a 5:3b 6:4a 7:4b 8:fr 9:p3 10:p2 11:w3 12:w2 13:w1 14:out
  const float *W[15], *R[15], *B[15];
  for (int i = 0; i < 15; ++i) {
    W[i] = (const float*)d_in[11 + 3 * i];
    R[i] = (const float*)d_in[12 + 3 * i];
    B[i] = (const float*)d_in[13 + 3 * i];
  }
  const float *g[4], *be[4];
  for (int i = 0; i < 4; ++i) { g[i] = (const float*)d_in[56 + 2 * i]; be[i] = (const float*)d_in[57 + 2 * i]; }

  int N[5]; N[0] = in_sizes[0];
  for (int i = 1; i < 5; ++i) N[i] = N[i - 1] / 2;      // 65536,32768,16384,8192,4096
  int E[5]; for (int i = 0; i < 5; ++i) E[i] = in_sizes[2 + i] / 2;
  int C[5] = {1, 32, 64, 128, 256};                      // channels of x at each level

  // ---- workspace bump allocator
  char* base = (char*)d_ws; size_t off = 0;
  auto alloc = [&](size_t elems) -> float* {
    float* p = (float*)(base + off);
    off += ((elems * sizeof(float) + 255) & ~(size_t)255);
    return p;
  };
  float* attr[5]; for (int l = 0; l < 5; ++l) attr[l] = alloc((size_t)E[l] * 2);
  float* inv[5];  for (int l = 0; l < 5; ++l) inv[l] = alloc((size_t)N[l]);
  float* ccnt[4]; for (int l = 0; l < 4; ++l) ccnt[l] = alloc((size_t)N[l + 1]);
  float* pos[5]; pos[0] = (float*)pos1; for (int l = 1; l < 5; ++l) pos[l] = alloc((size_t)N[l] * 2);
  float* wt[4]; for (int l = 0; l < 4; ++l) wt[l] = alloc((size_t)N[l] * C[l]);
  float* xs[5]; xs[0] = (float*)x; for (int l = 1; l < 5; ++l) xs[l] = alloc((size_t)N[l] * C[l]);
  size_t hmax = (size_t)N[0] * 32;
  float* hA = alloc(hmax);
  float* hB = alloc(hmax);
  float* gbuf = alloc((size_t)N[1] * 32);
  unsigned* segbuf = (unsigned*)alloc((size_t)N[1] * 32);
  float* psum = alloc((size_t)N[1] * 2);
  float* mean = alloc(256);
  float* istd = alloc(256);
  unsigned* maxu = (unsigned*)alloc(8);
  float* wpadA = alloc((size_t)1184 * 128);   // >= max KA4*CoutP (1152*128)
  float* wpadB = alloc((size_t)256 * 256);    // >= max Cin*CoutP
  float* acc = alloc((size_t)N[0] * 25 * 32); // >= max N*KA4 (65536*800)

  auto run_cart = [&](int l) {
    fill_u32<<<1, 32, 0, stream>>>(maxu, 0u, 1);
    edge_absmax<<<gdiv(E[l]), 256, 0, stream>>>(pos[l], ei[l], ei[l] + E[l], E[l], maxu);
    edge_cart<<<gdiv(E[l]), 256, 0, stream>>>(pos[l], ei[l], ei[l] + E[l], E[l], maxu, attr[l]);
  };
  auto run_indeg = [&](int l) {
    fill_f32<<<gdiv(N[l]), 256, 0, stream>>>(inv[l], 0.f, N[l]);
    count_idx<<<gdiv(E[l]), 256, 0, stream>>>(ei[l] + E[l], E[l], inv[l]);
    invert_cnt<<<gdiv(N[l]), 256, 0, stream>>>(inv[l], N[l]);
  };
  auto run_ccnt = [&](int l) {
    fill_f32<<<gdiv(N[l + 1]), 256, 0, stream>>>(ccnt[l], 0.f, N[l + 1]);
    count_idx<<<gdiv(N[l]), 256, 0, stream>>>(cl[l], N[l], ccnt[l]);
  };
  auto run_bn = [&](float* h, int Nn, int Cc, const float* gg, const float* bb) {
    bn_stats<<<Cc, 256, 0, stream>>>(h, Nn, Cc, mean, istd);
    bn_apply<<<gdiv((size_t)Nn * Cc), 256, 0, stream>>>(h, Nn, Cc, mean, istd, gg, bb);
  };
  auto run_pweights = [&](const float* xp, int l) {
    int ncl = N[l + 1], Cc = C[l];
    fill_f32<<<gdiv((size_t)ncl * Cc), 256, 0, stream>>>(gbuf, 0.f, (size_t)ncl * Cc);
    seg_sum_feat<<<gdiv((size_t)N[l] * Cc), 256, 0, stream>>>(xp, cl[l], N[l], Cc, gbuf);
    pweights_k<<<gdiv((size_t)N[l] * Cc), 256, 0, stream>>>(xp, cl[l], ccnt[l], gbuf, N[l], Cc, wt[l]);
  };
  auto run_pool = [&](const float* h, int l, int Cc) {
    int ncl = N[l + 1];
    fill_u32<<<gdiv((size_t)ncl * Cc), 256, 0, stream>>>(segbuf, ORD_NEG_INF, (size_t)ncl * Cc);
    seg_max_feat<<<gdiv((size_t)N[l] * Cc), 256, 0, stream>>>(h, cl[l], N[l], Cc, segbuf);
    segmax_decode<<<gdiv((size_t)ncl * Cc), 256, 0, stream>>>(segbuf, (size_t)ncl * Cc, xs[l + 1]);
    fill_f32<<<gdiv((size_t)ncl * 2), 256, 0, stream>>>(psum, 0.f, (size_t)ncl * 2);
    seg_sum_feat<<<gdiv((size_t)N[l] * 2), 256, 0, stream>>>(pos[l], cl[l], N[l], 2, psum);
    pos_div<<<gdiv((size_t)ncl * 2), 256, 0, stream>>>(psum, ccnt[l], ncl, pos[l + 1]);
  };

  // ================= encoder =================
  run_cart(0); run_indeg(0); run_ccnt(0);
  run_conv(stream, x, N[0], 1, ei[0], E[0], attr[0], inv[0], W[0], 25, 5, R[0], B[0], 32, nullptr, hA, 1, acc, wpadA, wpadB);
  run_conv(stream, hA, N[0], 32, ei[0], E[0], attr[0], inv[0], W[1], 25, 5, R[1], B[1], 32, nullptr, hB, 1, acc, wpadA, wpadB);
  run_bn(hB, N[0], 32, g[0], be[0]);
  run_pweights(x, 0);
  run_pool(hB, 0, 32);

  run_cart(1); run_indeg(1); run_ccnt(1);
  run_conv(stream, xs[1], N[1], 32, ei[1], E[1], attr[1], inv[1], W[2], 9, 3, R[2], B[2], 64, nullptr, hA, 1, acc, wpadA, wpadB);
  run_conv(stream, hA, N[1], 64, ei[1], E[1], attr[1], inv[1], W[3], 9, 3, R[3], B[3], 64, nullptr, hB, 1, acc, wpadA, wpadB);
  run_bn(hB, N[1], 64, g[1], be[1]);
  run_pweights(xs[1], 1);
  run_pool(hB, 1, 64);

  run_cart(2); run_indeg(2); run_ccnt(2);
  run_conv(stream, xs[2], N[2], 64, ei[2], E[2], attr[2], inv[2], W[4], 9, 3, R[4], B[4], 128, nullptr, hA, 1, acc, wpadA, wpadB);
  run_conv(stream, hA, N[2], 128, ei[2], E[2], attr[2], inv[2], W[5], 9, 3, R[5], B[5], 128, nullptr, hB, 1, acc, wpadA, wpadB);
  run_bn(hB, N[2], 128, g[2], be[2]);
  run_pweights(xs[2], 2);
  run_pool(hB, 2, 128);

  run_cart(3); run_indeg(3); run_ccnt(3);
  run_conv(stream, xs[3], N[3], 128, ei[3], E[3], attr[3], inv[3], W[6], 1, 1, R[6], B[6], 256, nullptr, hA, 1, acc, wpadA, wpadB);
  run_conv(stream, hA, N[3], 256, ei[3], E[3], attr[3], inv[3], W[7], 1, 1, R[7], B[7], 256, nullptr, hB, 1, acc, wpadA, wpadB);
  run_bn(hB, N[3], 256, g[3], be[3]);
  run_pweights(xs[3], 3);
  run_pool(hB, 3, 256);

  run_cart(4); run_indeg(4);
  run_conv(stream, xs[4], N[4], 256, ei[4], E[4], attr[4], inv[4], W[8], 1, 1, R[8], B[8], 128, nullptr, hB, 1, acc, wpadA, wpadB);

  // ================= decoder =================
  upsample_mul<<<gdiv((size_t)N[3] * 128), 256, 0, stream>>>(hB, cl[3], wt[3], N[3], 128, hA);
  run_conv(stream, xs[3], N[3], 128, ei[3], E[3], attr[3], inv[3], W[9], 9, 3, R[9], B[9], 128, hA, hB, 1, acc, wpadA, wpadB);
  run_conv(stream, hB, N[3], 128, ei[3], E[3], attr[3], inv[3], W[11], 1, 1, R[11], B[11], 64, nullptr, hA, 1, acc, wpadA, wpadB);
  upsample_mul<<<gdiv((size_t)N[2] * 64), 256, 0, stream>>>(hA, cl[2], wt[2], N[2], 64, hB);
  run_conv(stream, xs[2], N[2], 64, ei[2], E[2], attr[2], inv[2], W[10], 9, 3, R[10], B[10], 64, hB, hA, 1, acc, wpadA, wpadB);
  run_conv(stream, hA, N[2], 64, ei[2], E[2], attr[2], inv[2], W[12], 1, 1, R[12], B[12], 32, nullptr, hB, 1, acc, wpadA, wpadB);
  upsample_mul<<<gdiv((size_t)N[1] * 32), 256, 0, stream>>>(hB, cl[1], wt[1], N[1], 32, hA);
  run_conv(stream, hA, N[1], 32, ei[1], E[1], attr[1], inv[1], W[13], 1, 1, R[13], B[13], 1, nullptr, hB, 1, acc, wpadA, wpadB);
  upsample_mul<<<gdiv((size_t)N[0] * 1), 256, 0, stream>>>(hB, cl[0], wt[0], N[0], 1, hA);
  run_conv(stream, hA, N[0], 1, ei[0], E[0], attr[0], inv[0], W[14], 25, 5, R[14], B[14], 1, nullptr,
           (float*)d_out, 0, acc, wpadA, wpadB);
}